// ListMLECriterion_31464930411239
// MI455X (gfx1250) — compile-verified
//
#include <hip/hip_runtime.h>
#include <stdint.h>

#define ROW_L   2048
#define THREADS 256
#define PER_T   (ROW_L / THREADS)
#define NSEG    (ROW_L / 16)          // 128 segments of 16

#if defined(__HIP_DEVICE_COMPILE__) && defined(__gfx1250__) && \
    __has_builtin(__builtin_amdgcn_tensor_load_to_lds) && \
    __has_builtin(__builtin_amdgcn_s_wait_tensorcnt)
#define USE_TDM 1
#else
#define USE_TDM 0
#endif

#if defined(__HIP_DEVICE_COMPILE__) && defined(__gfx1250__) && \
    __has_builtin(__builtin_amdgcn_wmma_f32_16x16x4_f32)
#define USE_WMMA 1
#else
#define USE_WMMA 0
#endif

typedef __attribute__((ext_vector_type(4))) unsigned int uint32x4;
typedef __attribute__((ext_vector_type(8))) int          int32x8;
typedef __attribute__((ext_vector_type(4))) int          int32x4;
typedef __attribute__((ext_vector_type(2))) float        floatx2;
typedef __attribute__((ext_vector_type(8))) float        floatx8;

struct Smem {
    union {
        struct { float pred[ROW_L]; float label[ROW_L]; } st;  // staging (16 KB)
        float scan[ROW_L];                                     // reused after sort (8 KB)
    } u;
    unsigned long long kv[ROW_L];   // sort keys: [63:32]=desc-label key, [31:0]=pred bits
    float seg[NSEG];                // segment totals -> inclusive suffix sums
    float red[THREADS];
};

// Issue one TDM load of `nelts` contiguous f32 from global `gaddr` into LDS byte
// offset `lds_off`. D# layout per CDNA5 ISA §8.3-8.6 (2D descriptor, groups 2/3 zero).
__device__ __forceinline__ void tdm_load_row_f32(uint64_t gaddr, uint32_t lds_off,
                                                 uint32_t nelts) {
#if USE_TDM
    uint32x4 g0;
    g0[0] = 1u;                                          // count=1 (valid user D#)
    g0[1] = lds_off;                                     // lds_addr (bytes)
    g0[2] = (uint32_t)(gaddr & 0xFFFFFFFFull);           // global_addr[31:0]
    g0[3] = (uint32_t)((gaddr >> 32) & 0x01FFFFFFull)    // global_addr[56:32]
          | 0x80000000u;                                 // type=2 ("image")
    int32x8 g1;
    g1[0] = (int)(2u << 16);                             // workgroup_mask=0, data_size=4B
    g1[1] = (int)((nelts & 0xFFFFu) << 16);              // tensor_dim0[15:0]
    g1[2] = (int)(((nelts >> 16) & 0xFFFFu) | (1u << 16)); // tensor_dim0[31:16], tensor_dim1=1
    g1[3] = (int)((nelts & 0xFFFFu) << 16);              // tile_dim0 = nelts
    g1[4] = 1;                                           // tile_dim1=1, tile_dim2=0
    g1[5] = (int)nelts;                                  // tensor_dim0_stride[31:0]
    g1[6] = 0;
    g1[7] = 0;
    int32x4 z4 = {0, 0, 0, 0};
    int32x8 z8 = {0, 0, 0, 0, 0, 0, 0, 0};
    __builtin_amdgcn_tensor_load_to_lds(g0, g1, z4, z4, z8, 0);
#else
    (void)gaddr; (void)lds_off; (void)nelts;
#endif
}

__global__ __launch_bounds__(THREADS)
void listmle_row_kernel(const float* __restrict__ preds,
                        const float* __restrict__ labels,
                        float* __restrict__ rowsum) {
    __shared__ Smem s;
    const int row = blockIdx.x;
    const int tid = threadIdx.x;

#if USE_TDM
    if (tid < 32) {   // wave 0 only: TDM issues per-wave, EXEC ignored
        tdm_load_row_f32((uint64_t)(uintptr_t)(preds + (size_t)row * ROW_L), 0u, ROW_L);
        tdm_load_row_f32((uint64_t)(uintptr_t)(labels + (size_t)row * ROW_L),
                         (uint32_t)(ROW_L * sizeof(float)), ROW_L);
        __builtin_amdgcn_s_wait_tensorcnt(0);
    }
    __syncthreads();
#else
    for (int i = tid; i < ROW_L; i += THREADS) {
        s.u.st.pred[i]  = preds[(size_t)row * ROW_L + i];
        s.u.st.label[i] = labels[(size_t)row * ROW_L + i];
    }
    __syncthreads();
#endif

    // Build u64 sort keys (descending label order via inverted monotone map) + row max.
    float lmax = -__builtin_inff();
    for (int i = tid; i < ROW_L; i += THREADS) {
        float pr = s.u.st.pred[i];
        float lb = s.u.st.label[i];
        uint32_t b  = __float_as_uint(lb);
        uint32_t ka = (b & 0x80000000u) ? ~b : (b | 0x80000000u);  // ascending float key
        uint32_t kd = ~ka;                                         // descending
        s.kv[i] = ((uint64_t)kd << 32) | (uint64_t)__float_as_uint(pr);
        lmax = fmaxf(lmax, pr);
    }
    s.red[tid] = lmax;
    __syncthreads();
    for (int st = THREADS / 2; st > 0; st >>= 1) {
        if (tid < st) s.red[tid] = fmaxf(s.red[tid], s.red[tid + st]);
        __syncthreads();
    }
    const float M = s.red[0];
    __syncthreads();

    // LDS bitonic sort, ascending on u64 key (=> descending label).
    for (unsigned k = 2; k <= ROW_L; k <<= 1) {
        for (unsigned j = k >> 1; j > 0; j >>= 1) {
            for (unsigned i = tid; i < ROW_L; i += THREADS) {
                unsigned ixj = i ^ j;
                if (ixj > i) {
                    unsigned long long a = s.kv[i];
                    unsigned long long b = s.kv[ixj];
                    bool up = ((i & k) == 0);
                    if ((a > b) == up) { s.kv[i] = b; s.kv[ixj] = a; }
                }
            }
            __syncthreads();
        }
    }

    // exp(x - M) into scan buffer (aliases staging; stage data no longer needed).
    for (int i = tid; i < ROW_L; i += THREADS) {
        float x = __uint_as_float((uint32_t)(s.kv[i] & 0xFFFFFFFFull));
        s.u.scan[i] = __expf(x - M);
    }
    __syncthreads();

#if USE_WMMA
    // ---- Within-segment suffix sums via WMMA: D = U(16x16, upper-tri ones) * X ----
    // Each wave owns 16 segments (256 contiguous floats). X[r][c] = scan[g + 16c + r].
    // V_WMMA_F32_16X16X4_F32 chained 4x over K.
    {
        const int lane = tid & 31;
        const int wv   = tid >> 5;
        const int g    = wv * 256;
        const int m    = lane & 15;      // A row (M) == B/D column (N)
        const int hi   = lane >> 4;      // upper half-lanes carry K+2 / rows M+8
        floatx8 acc = {0.f, 0.f, 0.f, 0.f, 0.f, 0.f, 0.f, 0.f};
#pragma unroll
        for (int k0 = 0; k0 < 16; k0 += 4) {
            const int kb = k0 + (hi << 1);
            floatx2 A, B;
            A.x = (kb     >= m) ? 1.0f : 0.0f;   // U[m][kb]
            A.y = (kb + 1 >= m) ? 1.0f : 0.0f;   // U[m][kb+1]
            B.x = s.u.scan[g + 16 * m + kb];     // X[kb][m]
            B.y = s.u.scan[g + 16 * m + kb + 1]; // X[kb+1][m]
            acc = __builtin_amdgcn_wmma_f32_16x16x4_f32(
                false, A, false, B, (short)0, acc, false, false);
        }
        const int rb = hi << 3;                  // rows 0..7 / 8..15
#pragma unroll
        for (int v = 0; v < 8; ++v)
            s.u.scan[g + 16 * m + rb + v] = acc[v];
    }
    __syncthreads();

    // Segment totals = within-suffix at r=0; inclusive suffix-scan 128 of them.
    if (tid < NSEG) s.seg[tid] = s.u.scan[tid * 16];
    __syncthreads();
    for (int off = 1; off < NSEG; off <<= 1) {
        float v = 0.0f;
        if (tid < NSEG) v = s.seg[tid] + ((tid + off < NSEG) ? s.seg[tid + off] : 0.0f);
        __syncthreads();
        if (tid < NSEG) s.seg[tid] = v;
        __syncthreads();
    }

    // Row total of (log(tail) + M - x): tail = within[i] + suffix of later segments.
    float part = 0.0f;
    for (int i = tid; i < ROW_L; i += THREADS) {
        float x = __uint_as_float((uint32_t)(s.kv[i] & 0xFFFFFFFFull));
        int sidx = i >> 4;
        float tail = s.u.scan[i] + ((sidx < NSEG - 1) ? s.seg[sidx + 1] : 0.0f);
        part += __logf(tail) + M - x;
    }
#else
    // Fallback: full-array suffix inclusive scan (Hillis-Steele, 11 passes).
    for (int off = 1; off < ROW_L; off <<= 1) {
        float v[PER_T];
#pragma unroll
        for (int q = 0; q < PER_T; ++q) {
            int i = tid + q * THREADS;
            int j = i + off;
            v[q] = s.u.scan[i] + ((j < ROW_L) ? s.u.scan[j] : 0.0f);
        }
        __syncthreads();
#pragma unroll
        for (int q = 0; q < PER_T; ++q) s.u.scan[tid + q * THREADS] = v[q];
        __syncthreads();
    }
    float part = 0.0f;
    for (int i = tid; i < ROW_L; i += THREADS) {
        float x = __uint_as_float((uint32_t)(s.kv[i] & 0xFFFFFFFFull));
        part += __logf(s.u.scan[i]) + M - x;
    }
#endif

    s.red[tid] = part;
    __syncthreads();
    for (int st = THREADS / 2; st > 0; st >>= 1) {
        if (tid < st) s.red[tid] += s.red[tid + st];
        __syncthreads();
    }
    if (tid == 0) rowsum[row] = s.red[0];
}

__global__ __launch_bounds__(THREADS)
void listmle_reduce_kernel(const float* __restrict__ rows, float* __restrict__ out, int n) {
    __shared__ float red[THREADS];
    float p = 0.0f;
    for (int i = threadIdx.x; i < n; i += THREADS) p += rows[i];
    red[threadIdx.x] = p;
    __syncthreads();
    for (int st = THREADS / 2; st > 0; st >>= 1) {
        if (threadIdx.x < st) red[threadIdx.x] += red[threadIdx.x + st];
        __syncthreads();
    }
    if (threadIdx.x == 0) out[0] = red[0];
}

extern "C" void kernel_launch(void* const* d_in, const int* in_sizes, int n_in,
                              void* d_out, int out_size, void* d_ws, size_t ws_size,
                              hipStream_t stream) {
    const float* preds  = (const float*)d_in[0];
    const float* labels = (const float*)d_in[1];
    const int total = in_sizes[0];
    const int B = total / ROW_L;              // 8192 rows
    float* rowsum = (float*)d_ws;             // B floats of scratch

    listmle_row_kernel<<<B, THREADS, 0, stream>>>(preds, labels, rowsum);
    listmle_reduce_kernel<<<1, THREADS, 0, stream>>>(rowsum, (float*)d_out, B);
}